// MultiHeadSelfAttention_2954937499870
// MI455X (gfx1250) — compile-verified
//
#include <hip/hip_runtime.h>
#include <hip/hip_bf16.h>

// ---------------------------------------------------------------------------
// MultiHeadSelfAttention forward, fp32, for gfx1250 (MI455X).
//   B=2, S=2048, D=1024, H=16, DK=64, causal, RoPE(theta=1e4) on Q,K.
// Matmuls: V_WMMA_F32_16X16X4_F32.  LDS staging: GLOBAL_LOAD_ASYNC_TO_LDS_B128
// (ASYNCcnt), software-pipelined (triple buffer) in the projection GEMM.
// ---------------------------------------------------------------------------

typedef __attribute__((ext_vector_type(2))) float v2f;
typedef __attribute__((ext_vector_type(8))) float v8f;

#define DMODEL 1024
#define NHEADS 16
#define DK     64
#define SEQ    2048
#define BATCH  2
#define MROWS  (BATCH * SEQ)          // 4096
#define LN_THETA 9.210340371976184f   // ln(10000)

static __device__ __forceinline__ v8f wmma4(v2f a, v2f b, v8f c) {
  return __builtin_amdgcn_wmma_f32_16x16x4_f32(false, a, false, b, (short)0, c,
                                               false, false);
}

// gfx1250 async global->LDS copy, 16B per lane.  LDS dest address (byte
// offset, low 32 bits of the flat address) in a VGPR, 64-bit global address
// in a VGPR pair.  Tracked by ASYNCcnt.
static __device__ __forceinline__ void async_ld_b128(unsigned lds_off,
                                                     const void* gptr) {
  asm volatile("global_load_async_to_lds_b128 %0, %1, off"
               :
               : "v"(lds_off), "v"((unsigned long long)(size_t)gptr)
               : "memory");
}
#define WAIT_ASYNC_0() asm volatile("s_wait_asynccnt 0x0" ::: "memory")
#define WAIT_ASYNC_3() asm volatile("s_wait_asynccnt 0x3" ::: "memory")

static __device__ __forceinline__ unsigned lds_base_u32(const void* p) {
  // Flat->LDS mapping is defined as truncation to the low 32 bits.
  return (unsigned)(size_t)p;
}

// ---------------------------------------------------------------------------
// GEMM: out = A(4096x1024, row-major) @ W^T (W is 1024x1024 row-major (N,K)).
// Block tile 64(M) x 128(N), 256 threads = 8 waves, each wave does 16x64.
// Triple-buffered async pipeline over 64 K-chunks of 16.
// mode: 0 = plain store out[m*1024+n] (Wo projection -> final output)
//       1 = RoPE + 1/sqrt(DK), head-split [B,H,S,DK]   (Q)
//       2 = RoPE, head-split                           (K)
//       3 = head-split only                            (V)
// ---------------------------------------------------------------------------
#define AS_STRIDE 20           // 64 rows x 16 k, float4-aligned, low conflict
#define WS_STRIDE 20           // 128 rows x 16 k
#define CS_STRIDE 132          // 64 rows x 128 n (epilogue bounce, overlaid)
#define AS_FLOATS (64 * AS_STRIDE)            // 1280
#define WS_FLOATS (128 * WS_STRIDE)           // 2560
#define BUF_FLOATS (AS_FLOATS + WS_FLOATS)    // 3840 per pipeline stage
#define SMEM_FLOATS (3 * BUF_FLOATS)          // 11520 > 64*CS_STRIDE = 8448

__global__ __launch_bounds__(256) void proj_kernel(
    const float* __restrict__ A, const float* __restrict__ W,
    float* __restrict__ out, const long long* __restrict__ pos, int mode) {
  __shared__ float smem[SMEM_FLOATS];

  const int tid  = threadIdx.x;
  const int lane = tid & 31;
  const int w    = tid >> 5;
  const int half = lane >> 4;   // K-pair select for fp32 WMMA fragments
  const int lr   = lane & 15;

  const int m0 = blockIdx.y * 64;
  const int n0 = blockIdx.x * 128;
  const int m_off = (w & 3) * 16;
  const int n_off = (w >> 2) * 64;

  v8f c0 = {}, c1 = {}, c2 = {}, c3 = {};

  // per-thread staging coordinates (64 rows x 16 k per tile, float4 each)
  const int arow = tid >> 2;          // 0..63
  const int acol = (tid & 3) * 4;     // 0,4,8,12
  const unsigned sbase = lds_base_u32(smem);
  const unsigned a_off = (unsigned)(arow * AS_STRIDE + acol) * 4u;
  const unsigned w_off0 = (unsigned)(AS_FLOATS + arow * WS_STRIDE + acol) * 4u;
  const unsigned w_off1 =
      (unsigned)(AS_FLOATS + (arow + 64) * WS_STRIDE + acol) * 4u;
  const float* Arow  = A + (size_t)(m0 + arow) * DMODEL + acol;
  const float* Wrow0 = W + (size_t)(n0 + arow) * DMODEL + acol;
  const float* Wrow1 = W + (size_t)(n0 + arow + 64) * DMODEL + acol;

  auto issue = [&](int buf, int kb) {
    const unsigned b = sbase + (unsigned)(buf * BUF_FLOATS) * 4u;
    async_ld_b128(b + a_off,  Arow + kb);
    async_ld_b128(b + w_off0, Wrow0 + kb);
    async_ld_b128(b + w_off1, Wrow1 + kb);
  };

  issue(0, 0);
  issue(1, 16);

  for (int t = 0; t < 64; ++t) {
    if (t == 63) { WAIT_ASYNC_0(); } else { WAIT_ASYNC_3(); }
    __syncthreads();                       // tile t staged for all waves
    if (t + 2 < 64) issue((t + 2) % 3, (t + 2) * 16);  // overlap with compute

    const float* As = smem + (t % 3) * BUF_FLOATS;
    const float* Ws = As + AS_FLOATS;
#pragma unroll
    for (int kk = 0; kk < 4; ++kk) {
      const int ko = kk * 4 + 2 * half;
      v2f a  = *(const v2f*)&As[(m_off + lr) * AS_STRIDE + ko];
      v2f b0 = *(const v2f*)&Ws[(n_off +  0 + lr) * WS_STRIDE + ko];
      v2f b1 = *(const v2f*)&Ws[(n_off + 16 + lr) * WS_STRIDE + ko];
      v2f b2 = *(const v2f*)&Ws[(n_off + 32 + lr) * WS_STRIDE + ko];
      v2f b3 = *(const v2f*)&Ws[(n_off + 48 + lr) * WS_STRIDE + ko];
      c0 = wmma4(a, b0, c0);
      c1 = wmma4(a, b1, c1);
      c2 = wmma4(a, b2, c2);
      c3 = wmma4(a, b3, c3);
    }
  }

  // ---- epilogue: C tiles -> LDS (overlaid) so RoPE column-pairs are easy ---
  __syncthreads();
  float* Cs = smem;
#pragma unroll
  for (int r = 0; r < 8; ++r) {
    const int row = m_off + r + 8 * half;
    Cs[row * CS_STRIDE + n_off +  0 + lr] = c0[r];
    Cs[row * CS_STRIDE + n_off + 16 + lr] = c1[r];
    Cs[row * CS_STRIDE + n_off + 32 + lr] = c2[r];
    Cs[row * CS_STRIDE + n_off + 48 + lr] = c3[r];
  }
  __syncthreads();

  for (int p = tid; p < 64 * 128 / 2; p += 256) {
    const int e  = p * 2;
    const int lm = e >> 7;         // local row 0..63
    const int lc = e & 127;        // local col (even)
    float x1 = Cs[lm * CS_STRIDE + lc];
    float x2 = Cs[lm * CS_STRIDE + lc + 1];
    const int gm = m0 + lm;        // global row  = b*SEQ + s
    const int gn = n0 + lc;        // global col  = h*DK + dk (dk even)
    const int b  = gm >> 11;
    const int s  = gm & (SEQ - 1);
    const int h  = gn >> 6;
    const int dk = gn & (DK - 1);

    if (mode == 1 || mode == 2) {
      const float fpos = (float)pos[s];
      const float freq = __expf(-(float)dk * (LN_THETA / (float)DK));
      float sn, cs;
      __sincosf(fpos * freq, &sn, &cs);
      float r1 = x1 * cs - x2 * sn;
      float r2 = x1 * sn + x2 * cs;
      if (mode == 1) { r1 *= 0.125f; r2 *= 0.125f; }  // 1/sqrt(DK)
      x1 = r1; x2 = r2;
    }
    float2 v; v.x = x1; v.y = x2;
    if (mode == 0) {
      *(float2*)&out[(size_t)gm * DMODEL + gn] = v;
    } else {
      const size_t addr = ((((size_t)b * NHEADS + h) * SEQ + s) * DK) + dk;
      *(float2*)&out[addr] = v;
    }
  }
}

// ---------------------------------------------------------------------------
// Flash attention, causal.  Q,K,V in [B,H,S,DK] (scale folded into Q).
// 128 threads (4 waves), 64 queries / block, 64-key LDS tiles staged with
// async global->LDS copies.  Output written into the (4096 x 1024) matrix
// consumed by the Wo projection.
// ---------------------------------------------------------------------------
#define KV_STRIDE 68   // 64 x 64 tile, float4-aligned, conflict-padded
#define P_STRIDE  66   // 16 x 64 probs per wave

__global__ __launch_bounds__(128) void attn_kernel(
    const float* __restrict__ Q, const float* __restrict__ K,
    const float* __restrict__ V, float* __restrict__ out) {
  __shared__ float Ks[64 * KV_STRIDE];
  __shared__ float Vs[64 * KV_STRIDE];
  __shared__ float Ps[4 * 16 * P_STRIDE];

  const int tid  = threadIdx.x;
  const int lane = tid & 31;
  const int w    = tid >> 5;
  const int half = lane >> 4;
  const int lr   = lane & 15;

  const int bh = blockIdx.y;            // b*NHEADS + h
  const int qb = blockIdx.x * 64;
  const int q0 = qb + w * 16;

  const float* Qp = Q + (size_t)bh * SEQ * DK;
  const float* Kp = K + (size_t)bh * SEQ * DK;
  const float* Vp = V + (size_t)bh * SEQ * DK;

  // Q A-fragments for the whole 16(q) x 64(dk) tile, kept in registers.
  v2f qa[16];
#pragma unroll
  for (int kk = 0; kk < 16; ++kk)
    qa[kk] = *(const v2f*)&Qp[(size_t)(q0 + lr) * DK + kk * 4 + 2 * half];

  v8f o0 = {}, o1 = {}, o2 = {}, o3 = {};
  float mrun[8], lrun[8];
#pragma unroll
  for (int r = 0; r < 8; ++r) { mrun[r] = -1e30f; lrun[r] = 0.0f; }

  float* Pw = &Ps[w * 16 * P_STRIDE];
  const unsigned kbase = lds_base_u32(Ks);
  const unsigned vbase = lds_base_u32(Vs);

  for (int kb = 0; kb <= qb; kb += 64) {
    __syncthreads();   // all waves done reading previous K/V tiles
    // stage 64x64 K and V tiles: async global->LDS, 16B per lane
#pragma unroll
    for (int i = 0; i < 4; ++i) {
      const int slot = i * 128 + tid;          // 0..511 of 1024 float4 slots
      const int row  = slot >> 4;
      const int c4   = (slot & 15) << 2;
      const unsigned loff = (unsigned)(row * KV_STRIDE + c4) * 4u;
      async_ld_b128(kbase + loff, &Kp[(size_t)(kb + row) * DK + c4]);
      async_ld_b128(vbase + loff, &Vp[(size_t)(kb + row) * DK + c4]);
      const int slot2 = slot + 512;
      const int row2  = slot2 >> 4;
      const int c42   = (slot2 & 15) << 2;
      const unsigned loff2 = (unsigned)(row2 * KV_STRIDE + c42) * 4u;
      async_ld_b128(kbase + loff2, &Kp[(size_t)(kb + row2) * DK + c42]);
      async_ld_b128(vbase + loff2, &Vp[(size_t)(kb + row2) * DK + c42]);
    }
    WAIT_ASYNC_0();
    __syncthreads();

    // scores: S = Q (16x64) x K^T (64x64) -> 4 tiles of 16x16
    v8f c0 = {}, c1 = {}, c2 = {}, c3 = {};
#pragma unroll
    for (int kk = 0; kk < 16; ++kk) {
      const int ko = kk * 4 + 2 * half;
      v2f a  = qa[kk];
      v2f b0 = *(const v2f*)&Ks[( 0 + lr) * KV_STRIDE + ko];
      v2f b1 = *(const v2f*)&Ks[(16 + lr) * KV_STRIDE + ko];
      v2f b2 = *(const v2f*)&Ks[(32 + lr) * KV_STRIDE + ko];
      v2f b3 = *(const v2f*)&Ks[(48 + lr) * KV_STRIDE + ko];
      c0 = wmma4(a, b0, c0);
      c1 = wmma4(a, b1, c1);
      c2 = wmma4(a, b2, c2);
      c3 = wmma4(a, b3, c3);
    }

    // causal mask: only the diagonal key block can exceed the query index
    if (kb == qb) {
#pragma unroll
      for (int r = 0; r < 8; ++r) {
        const int qi = q0 + r + 8 * half;
        if (kb +  0 + lr > qi) c0[r] = -1e30f;
        if (kb + 16 + lr > qi) c1[r] = -1e30f;
        if (kb + 32 + lr > qi) c2[r] = -1e30f;
        if (kb + 48 + lr > qi) c3[r] = -1e30f;
      }
    }

    // online softmax (rows live in 16-lane halves -> xor-shuffle reductions)
#pragma unroll
    for (int r = 0; r < 8; ++r) {
      float rm = fmaxf(fmaxf(c0[r], c1[r]), fmaxf(c2[r], c3[r]));
#pragma unroll
      for (int off = 8; off >= 1; off >>= 1)
        rm = fmaxf(rm, __shfl_xor(rm, off, 32));
      const float mn    = fmaxf(mrun[r], rm);
      const float alpha = __expf(mrun[r] - mn);
      const float p0 = __expf(c0[r] - mn);
      const float p1 = __expf(c1[r] - mn);
      const float p2 = __expf(c2[r] - mn);
      const float p3 = __expf(c3[r] - mn);
      float rs = p0 + p1 + p2 + p3;
#pragma unroll
      for (int off = 8; off >= 1; off >>= 1)
        rs += __shfl_xor(rs, off, 32);
      mrun[r] = mn;
      lrun[r] = lrun[r] * alpha + rs;

      const int prow = r + 8 * half;
      Pw[prow * P_STRIDE +  0 + lr] = p0;
      Pw[prow * P_STRIDE + 16 + lr] = p1;
      Pw[prow * P_STRIDE + 32 + lr] = p2;
      Pw[prow * P_STRIDE + 48 + lr] = p3;

      o0[r] *= alpha; o1[r] *= alpha; o2[r] *= alpha; o3[r] *= alpha;
    }

    // O += P (16x64 keys) x V (64 keys x 64 dk)
#pragma unroll
    for (int kk = 0; kk < 16; ++kk) {
      v2f a = *(const v2f*)&Pw[lr * P_STRIDE + kk * 4 + 2 * half];
      const int k0 = kk * 4 + 2 * half;
      v2f b0 = { Vs[k0 * KV_STRIDE +  0 + lr], Vs[(k0 + 1) * KV_STRIDE +  0 + lr] };
      v2f b1 = { Vs[k0 * KV_STRIDE + 16 + lr], Vs[(k0 + 1) * KV_STRIDE + 16 + lr] };
      v2f b2 = { Vs[k0 * KV_STRIDE + 32 + lr], Vs[(k0 + 1) * KV_STRIDE + 32 + lr] };
      v2f b3 = { Vs[k0 * KV_STRIDE + 48 + lr], Vs[(k0 + 1) * KV_STRIDE + 48 + lr] };
      o0 = wmma4(a, b0, o0);
      o1 = wmma4(a, b1, o1);
      o2 = wmma4(a, b2, o2);
      o3 = wmma4(a, b3, o3);
    }
  }

  // finalize: O / l, scatter to the (4096 x 1024) attn-out matrix
  const int b = bh >> 4;
  const int h = bh & 15;
#pragma unroll
  for (int r = 0; r < 8; ++r) {
    const float iv = 1.0f / lrun[r];
    const int gm = b * SEQ + q0 + r + 8 * half;
    float* orow = out + (size_t)gm * DMODEL + h * DK;
    orow[ 0 + lr] = o0[r] * iv;
    orow[16 + lr] = o1[r] * iv;
    orow[32 + lr] = o2[r] * iv;
    orow[48 + lr] = o3[r] * iv;
  }
}

// ---------------------------------------------------------------------------
extern "C" void kernel_launch(void* const* d_in, const int* in_sizes, int n_in,
                              void* d_out, int out_size, void* d_ws,
                              size_t ws_size, hipStream_t stream) {
  (void)in_sizes; (void)n_in; (void)out_size; (void)ws_size;
  const float* x  = (const float*)d_in[0];
  const float* Wq = (const float*)d_in[1];
  const float* Wk = (const float*)d_in[2];
  const float* Wv = (const float*)d_in[3];
  const float* Wo = (const float*)d_in[4];
  const long long* pos = (const long long*)d_in[5];

  float* ws = (float*)d_ws;
  const size_t NE = (size_t)MROWS * DMODEL;  // 4096*1024 elements per buffer
  float* Qb = ws;
  float* Kb = ws + NE;
  float* Vb = ws + 2 * NE;
  float* AO = ws + 3 * NE;

  dim3 gproj(DMODEL / 128, MROWS / 64);  // (8, 64)
  proj_kernel<<<gproj, 256, 0, stream>>>(x, Wq, Qb, pos, 1);
  proj_kernel<<<gproj, 256, 0, stream>>>(x, Wk, Kb, pos, 2);
  proj_kernel<<<gproj, 256, 0, stream>>>(x, Wv, Vb, pos, 3);

  dim3 gatt(SEQ / 64, BATCH * NHEADS);   // (32, 32)
  attn_kernel<<<gatt, 128, 0, stream>>>(Qb, Kb, Vb, AO);

  proj_kernel<<<gproj, 256, 0, stream>>>(AO, Wo, (float*)d_out, pos, 0);
}